// FixedActionDecoder_18150531792935
// MI455X (gfx1250) — compile-verified
//
#include <hip/hip_runtime.h>
#include <hip/hip_bf16.h>

typedef __attribute__((ext_vector_type(2))) float v2f;
typedef __attribute__((ext_vector_type(8))) float v8f;
typedef __attribute__((__vector_size__(4 * sizeof(int)))) int v4i;
typedef __attribute__((address_space(1))) v4i* gv4p;   // global (device) v4i*
typedef __attribute__((address_space(3))) v4i* lv4p;   // LDS (shared) v4i*

#define EMB_D    64
#define N_POINT  11
#define ROWS_PER_WAVE 16
#define WAVES_PER_BLK 8
#define ROWS_PER_BLK  (ROWS_PER_WAVE * WAVES_PER_BLK)   // 128
#define XROW_F   68     // staged row stride in floats (64 data + 4 pad) -> 272B
#define LDSB_STRIDE 66  // normalized B^T [n][k], padded for conflict-free b64 reads
#define LDSS_STRIDE 17  // sims [row][col], padded

#if defined(__has_builtin)
#  if __has_builtin(__builtin_amdgcn_global_load_async_to_lds_b128)
#    define USE_ASYNC_BUILTIN 1
#  else
#    define USE_ASYNC_BUILTIN 0
#  endif
#else
#  define USE_ASYNC_BUILTIN 0
#endif

__global__ __launch_bounds__(256) void
fixed_action_decoder_kernel(const float* __restrict__ x,     // [B, 64]
                            const float* __restrict__ av,    // [64, 11]
                            float* __restrict__ out,         // [B, 4]
                            int B) {
    __shared__ float ldsX[WAVES_PER_BLK * ROWS_PER_WAVE * XROW_F]; // 34816 B
    __shared__ float ldsB[16 * LDSB_STRIDE];                       //  4224 B
    __shared__ float ldsS[WAVES_PER_BLK][16 * LDSS_STRIDE];        //  8704 B

    const int tid  = threadIdx.x;
    const int wave = tid >> 5;
    const int lane = tid & 31;

    const long long rowBase = (long long)blockIdx.x * ROWS_PER_BLK +
                              (long long)wave * ROWS_PER_WAVE;
    float* slice = &ldsX[wave * ROWS_PER_WAVE * XROW_F];

    // ---- Async-stage this wave's 16 rows of x into LDS (coalesced 16B/lane,
    //      512B/instruction, tracked by ASYNCcnt). Rows clamped, not guarded,
    //      so no EXEC divergence; tail rows redundantly re-read row B-1. ----
    #pragma unroll
    for (int i = 0; i < 8; ++i) {
        const int f    = (i * 32 + lane) * 16;   // byte index within 4KB tile
        const int r16  = f >> 8;                 // row within tile
        const int colb = f & 255;                // byte within 256B row
        long long grow = rowBase + r16;
        if (grow > (long long)(B - 1)) grow = (long long)(B - 1);
        const char* gp = (const char*)x + grow * 256 + colb;
        char*       lp = (char*)slice + r16 * (XROW_F * 4) + colb;
#if USE_ASYNC_BUILTIN
        __builtin_amdgcn_global_load_async_to_lds_b128(
            (gv4p)(uintptr_t)gp, (lv4p)(uintptr_t)lp, 0, 0);
#else
        asm volatile("global_load_async_to_lds_b128 %0, %1, off"
                     :: "v"((unsigned)(uintptr_t)lp),
                        "v"((unsigned long long)(uintptr_t)gp)
                     : "memory");
#endif
    }

    // ---- Normalize action-vector columns into ldsB[n][k] (transposed),
    //      columns padded to 16 with zeros. Row-norm of x is skipped: it
    //      uniformly scales each row's sims and cannot change the argmax. ----
    if (tid < 16) {
        const int n = tid;
        if (n < N_POINT) {
            float s = 0.0f;
            #pragma unroll
            for (int k = 0; k < EMB_D; ++k) {
                const float v = av[k * N_POINT + n];
                s += v * v;
            }
            const float inv = 1.0f / fmaxf(sqrtf(s), 1e-8f);
            #pragma unroll
            for (int k = 0; k < EMB_D; ++k)
                ldsB[n * LDSB_STRIDE + k] = av[k * N_POINT + n] * inv;
        } else {
            #pragma unroll
            for (int k = 0; k < EMB_D; ++k)
                ldsB[n * LDSB_STRIDE + k] = 0.0f;
        }
    }
    __syncthreads();                                  // ldsB visible to block
    asm volatile("s_wait_asynccnt 0x0" ::: "memory"); // this wave's X tile in LDS

    const int nm   = lane & 15;          // A: tile row M; B: column N
    const int kgrp = (lane >> 4) << 1;   // 0 (lanes 0-15) or 2 (lanes 16-31)

    // ---- Preload all fragments, then 16 back-to-back chained WMMAs ----
    v2f bfrag[16];
    #pragma unroll
    for (int t = 0; t < 16; ++t)
        bfrag[t] = *(const v2f*)&ldsB[nm * LDSB_STRIDE + 4 * t + kgrp];

    v2f afrag[16];
    #pragma unroll
    for (int t = 0; t < 16; ++t)
        afrag[t] = *(const v2f*)(slice + nm * XROW_F + 4 * t + kgrp);

    v8f c = {};
    #pragma unroll
    for (int t = 0; t < 16; ++t)
        c = __builtin_amdgcn_wmma_f32_16x16x4_f32(
                false, afrag[t], false, bfrag[t],
                (short)0, c, false, false);

    // ---- Spill D tile: lane holds column N=nm, rows v + 8*(lane>=16) ----
    const int rhalf = (lane >> 4) * 8;
    #pragma unroll
    for (int v = 0; v < 8; ++v)
        ldsS[wave][(rhalf + v) * LDSS_STRIDE + nm] = c[v];
    __syncthreads();

    // ---- Segment-max over action groups + argmax + one-hot float4 store ----
    // groups: [0..3]->0, [4..8]->1, [9]->2, [10]->3
    if (lane < 16) {
        const long long orow = rowBase + lane;
        if (orow < (long long)B) {
            const float* s = &ldsS[wave][lane * LDSS_STRIDE];
            const float g0 = fmaxf(fmaxf(s[0], s[1]), fmaxf(s[2], s[3]));
            const float g1 = fmaxf(fmaxf(fmaxf(s[4], s[5]), fmaxf(s[6], s[7])), s[8]);
            const float g2 = s[9];
            const float g3 = s[10];
            int   idx  = 0;
            float best = g0;
            if (g1 > best) { best = g1; idx = 1; }   // strict '>' == first-max tie-break
            if (g2 > best) { best = g2; idx = 2; }
            if (g3 > best) { best = g3; idx = 3; }
            float4 o;
            o.x = (idx == 0) ? 1.0f : 0.0f;
            o.y = (idx == 1) ? 1.0f : 0.0f;
            o.z = (idx == 2) ? 1.0f : 0.0f;
            o.w = (idx == 3) ? 1.0f : 0.0f;
            *(float4*)(out + orow * 4) = o;
        }
    }
}

extern "C" void kernel_launch(void* const* d_in, const int* in_sizes, int n_in,
                              void* d_out, int out_size, void* d_ws, size_t ws_size,
                              hipStream_t stream) {
    const float* x  = (const float*)d_in[0];   // embedded_words [B, 64]
    const float* av = (const float*)d_in[1];   // action_vectors [64, 11]
    float* out = (float*)d_out;                // [B, 4] one-hot fp32

    const int B = in_sizes[0] / EMB_D;
    const int grid = (B + ROWS_PER_BLK - 1) / ROWS_PER_BLK;

    fixed_action_decoder_kernel<<<grid, 256, 0, stream>>>(x, av, out, B);
}